// WeightFusion_64630667870840
// MI455X (gfx1250) — compile-verified
//
#include <hip/hip_runtime.h>
#include <math.h>

// Problem constants (from reference): B=2048, N=128, D=512
#define BB   2048
#define NMAX 128
#define DD   512
#define ZR   516   // padded LDS row stride in floats (TDM pad: +2 DW every 256 DW)

typedef __attribute__((ext_vector_type(2))) float v2f;
typedef __attribute__((ext_vector_type(8))) float v8f;
typedef __attribute__((ext_vector_type(4))) unsigned int u32x4;
typedef __attribute__((ext_vector_type(8))) int i32x8;
typedef __attribute__((ext_vector_type(4))) int i32x4;

// TDM pad layout: element (n,d) lives at n*ZR + d + 2*(d>=256)
__device__ __forceinline__ int ZOFF(int d) { return d + ((d >> 8) << 1); }

// ---------------------------------------------------------------------------
// Kernel 1: CW[2B, D] = CONTENT[2B, D] @ cow[D, D] via v_wmma_f32_16x16x4_f32
// ---------------------------------------------------------------------------
__global__ __launch_bounds__(256)
void cw_gemm_kernel(const float* __restrict__ text,
                    const float* __restrict__ img,
                    const float* __restrict__ cow,
                    float* __restrict__ CW) {
  const int lane = threadIdx.x & 31;
  const int wave = threadIdx.x >> 5;
  const int tile = blockIdx.x * 8 + wave;   // 8192 tiles = (4096/16) * (512/16)
  const int tileRow = tile >> 5;
  const int tileCol = tile & 31;
  const int col = lane & 15;
  const int hf  = lane >> 4;

  const int r    = tileRow * 16 + col;
  const int bidx = r >> 1;
  const float* aRow = ((r & 1) ? img : text) + (size_t)bidx * DD;
  const int c = tileCol * 16 + col;

  v8f acc = {0.f,0.f,0.f,0.f,0.f,0.f,0.f,0.f};
  for (int k0 = 0; k0 < DD; k0 += 4) {
    const int ka = k0 + 2 * hf;
    v2f a = *(const v2f*)(aRow + ka);
    v2f b;
    b.x = cow[(size_t)ka * DD + c];
    b.y = cow[(size_t)(ka + 1) * DD + c];
    acc = __builtin_amdgcn_wmma_f32_16x16x4_f32(false, a, false, b,
                                                (short)0, acc, false, false);
  }
  #pragma unroll
  for (int i = 0; i < 8; ++i) {
    const int row = tileRow * 16 + hf * 8 + i;
    CW[(size_t)row * DD + c] = acc[i];
  }
}

// ---------------------------------------------------------------------------
// Kernel 2: one workgroup per sample. The whole masked z tile (128x512 f32)
// is DMA'd into LDS by ONE Tensor-Data-Mover op (tensor_load_to_lds):
//   - tensor_dim1 = cnum  -> rows n >= cnum read OOB and are written as zeros
//   - pad_enable, interval=256DW, amount=2DW -> row stride 516 (bank-friendly)
// All four attention phases then run out of LDS (single HBM pass).
// ---------------------------------------------------------------------------
__global__ __launch_bounds__(512)
void coattn_kernel(const float* __restrict__ text,
                   const float* __restrict__ img,
                   const float* __restrict__ comment,
                   const int*   __restrict__ comment_num,
                   const float* __restrict__ W_ca, const float* __restrict__ b_ca,
                   const float* __restrict__ W_co, const float* __restrict__ b_co,
                   const float* __restrict__ CW,
                   float* __restrict__ out) {
  extern __shared__ float sm[];
  float* zs  = sm;                 // [128][ZR]  masked comment rows (TDM dest)
  float* cw2 = zs  + NMAX * ZR;    // [2][ZR]    cw = content @ cow (from ws)
  float* ct  = cw2 + 2 * ZR;       // [2][ZR]    content rows (text,img)
  float* cwn = ct  + 2 * ZR;       // [2][128]   co_w
  float* wca = cwn + 2 * NMAX;     // [512]
  float* wco = wca + DD;           // [512]
  float* zl  = wco + DD;           // [128]      zw logits
  float* zw  = zl  + NMAX;         // [128]      comment softmax
  float* cwa = zw  + NMAX;         // [2]        content softmax
  float* red = cwa + 2;            // [16]       block-reduce scratch

  const int b    = blockIdx.x;
  const int tid  = threadIdx.x;    // 0..511 == D
  const int lane = tid & 31;
  const int wave = tid >> 5;       // 16 waves
  const int cnum = comment_num[b];

  // ---- Issue the TDM load of z early so the DMA overlaps Phase A ----------
  if (wave == 0) {
    const unsigned long long gaddr =
        (unsigned long long)(size_t)(comment + (size_t)b * NMAX * DD);
    u32x4 g0;
    g0[0] = 1u;                                              // count=1 (valid D#)
    g0[1] = (unsigned)__builtin_amdgcn_groupstaticsize();    // lds_addr of zs
    g0[2] = (unsigned)(gaddr & 0xFFFFFFFFu);                 // global_addr[31:0]
    g0[3] = (unsigned)((gaddr >> 32) & 0x01FFFFFFu)          // global_addr[56:32]
          | 0x80000000u;                                     // type=2 ("image")
    i32x8 g1;
    g1[0] = (int)((2u << 16)     // data_size = 4B
                | (1u << 20)     // pad_enable
                | (7u << 22)     // pad_interval: every 256 DWORDs
                | (1u << 25));   // pad_amount: 2 DWORDs
    g1[1] = (int)((unsigned)DD << 16);            // tensor_dim0[15:0] = 512
    g1[2] = (int)(((unsigned)cnum & 0xFFFFu) << 16); // tensor_dim1[15:0] = cnum (ragged mask!)
    g1[3] = (int)((unsigned)DD << 16);            // tile_dim0 = 512
    g1[4] = NMAX;                                 // tile_dim1 = 128 (tile_dim2 = 0)
    g1[5] = DD;                                   // tensor_dim0_stride = 512
    g1[6] = 0;
    g1[7] = 0;
    i32x4 g2 = {0, 0, 0, 0};
    i32x4 g3 = {0, 0, 0, 0};
    i32x8 g4 = {0, 0, 0, 0, 0, 0, 0, 0};          // unused group (6-arg toolchain form)
    __builtin_amdgcn_tensor_load_to_lds(g0, g1, g2, g3, g4, 0);
  }

  // ---- Phase A: stage small operands (overlaps the TDM DMA) ---------------
  ct[tid]        = text[(size_t)b * DD + tid];
  ct[ZR + tid]   = img [(size_t)b * DD + tid];
  cw2[tid]       = CW[((size_t)b * 2 + 0) * DD + tid];
  cw2[ZR + tid]  = CW[((size_t)b * 2 + 1) * DD + tid];
  wca[tid] = W_ca[tid];
  wco[tid] = W_co[tid];

  if (wave == 0) __builtin_amdgcn_s_wait_tensorcnt(0);       // z tile landed
  __syncthreads();

  // ---- Phase C: co_w = tanh(cw @ z^T) via WMMA ---------------------------
  if (wave < 8) {
    const int n0  = wave * 16;
    const int col = lane & 15;
    const int hf  = lane >> 4;
    const float* ap = cw2 + (col & 1) * ZR;                  // A row (replicated 0/1)
    const float* bp = zs + (n0 + col) * ZR;                  // B[k][n] = z[n][k]
    v8f acc = {0.f,0.f,0.f,0.f,0.f,0.f,0.f,0.f};
    for (int k0 = 0; k0 < 256; k0 += 4) {                    // first pad region
      const int ka = k0 + 2 * hf;
      v2f a  = *(const v2f*)(ap + ka);
      v2f bb = *(const v2f*)(bp + ka);
      acc = __builtin_amdgcn_wmma_f32_16x16x4_f32(false, a, false, bb,
                                                  (short)0, acc, false, false);
    }
    for (int k0 = 256; k0 < DD; k0 += 4) {                   // past the 2-DW pad
      const int ka = k0 + 2 * hf;
      v2f a  = *(const v2f*)(ap + ka);
      v2f bb = *(const v2f*)(bp + ka + 2);
      acc = __builtin_amdgcn_wmma_f32_16x16x4_f32(false, a, false, bb,
                                                  (short)0, acc, false, false);
    }
    if (hf == 0) {                                           // M=0 -> acc[0], M=1 -> acc[1]
      cwn[n0 + col]        = tanhf(acc[0]);
      cwn[NMAX + n0 + col] = tanhf(acc[1]);
    }
  }
  __syncthreads();

  // ---- Phase D: content attention ----------------------------------------
  const int tz = ZOFF(tid);
  #pragma unroll
  for (int k = 0; k < 2; ++k) {
    float acc = ct[k * ZR + tid];
    const float* cr = cwn + k * NMAX;
    #pragma unroll 8
    for (int n = 0; n < NMAX; ++n)
      acc += cr[n] * zs[n * ZR + tz];                        // co_w @ z
    float part = tanhf(acc) * wca[tid];
    for (int off = 16; off; off >>= 1) part += __shfl_down(part, off, 32);
    if (lane == 0) red[wave] = part;
    __syncthreads();
    if (tid == 0) {
      float s = 0.f;
      for (int w = 0; w < 16; ++w) s += red[w];
      cwa[k] = s;                                            // raw logit (pre-bias)
    }
    __syncthreads();
  }
  if (tid == 0) {
    const float l0 = cwa[0] + b_ca[0], l1 = cwa[1] + b_ca[0];
    const float m  = fmaxf(l0, l1);
    const float e0 = __expf(l0 - m), e1 = __expf(l1 - m);
    const float s  = e0 + e1;
    cwa[0] = e0 / s; cwa[1] = e1 / s;
    out[(size_t)2 * BB * DD + (size_t)b * 2 + 0] = cwa[0];
    out[(size_t)2 * BB * DD + (size_t)b * 2 + 1] = cwa[1];
  }
  __syncthreads();
  out[(size_t)b * DD + tid] = ct[tid] * cwa[0] + ct[ZR + tid] * cwa[1];

  // ---- Phase E: comment attention (masked softmax over N) ----------------
  for (int n = wave; n < NMAX; n += 16) {
    const float c0 = cwn[n], c1 = cwn[NMAX + n];
    float acc = 0.f;
    for (int d0 = lane; d0 < DD; d0 += 32)
      acc += tanhf(zs[n * ZR + ZOFF(d0)] + c0 * ct[d0] + c1 * ct[ZR + d0]) * wco[d0];
    for (int off = 16; off; off >>= 1) acc += __shfl_down(acc, off, 32);
    if (lane == 0) zl[n] = (n < cnum) ? (acc + b_co[0]) : -__builtin_inff();
  }
  __syncthreads();
  if (wave == 0) {
    float m = -__builtin_inff();
    #pragma unroll
    for (int j = 0; j < 4; ++j) m = fmaxf(m, zl[lane + 32 * j]);
    for (int off = 16; off; off >>= 1) m = fmaxf(m, __shfl_xor(m, off, 32));
    float e[4]; float s = 0.f;
    #pragma unroll
    for (int j = 0; j < 4; ++j) {
      const int n = lane + 32 * j;
      e[j] = (n < cnum) ? __expf(zl[n] - m) : 0.f;
      s += e[j];
    }
    for (int off = 16; off; off >>= 1) s += __shfl_xor(s, off, 32);
    #pragma unroll
    for (int j = 0; j < 4; ++j) zw[lane + 32 * j] = e[j] / s;
  }
  __syncthreads();

  // ---- Phase F: refine_comment = zw^T @ z --------------------------------
  {
    float acc = 0.f;
    #pragma unroll 8
    for (int n = 0; n < NMAX; ++n) acc += zw[n] * zs[n * ZR + tz];
    out[(size_t)BB * DD + (size_t)b * DD + tid] = acc;
  }
}

// ---------------------------------------------------------------------------
extern "C" void kernel_launch(void* const* d_in, const int* in_sizes, int n_in,
                              void* d_out, int out_size, void* d_ws, size_t ws_size,
                              hipStream_t stream) {
  const float* text        = (const float*)d_in[0];
  const float* img         = (const float*)d_in[1];
  const float* comment     = (const float*)d_in[2];
  const int*   comment_num = (const int*)  d_in[3];
  const float* cow         = (const float*)d_in[4];
  const float* W_ca        = (const float*)d_in[5];
  const float* b_ca        = (const float*)d_in[6];
  const float* W_co        = (const float*)d_in[7];
  const float* b_co        = (const float*)d_in[8];
  float* out = (float*)d_out;
  float* CW  = (float*)d_ws;   // [2B, D] f32 = 8 MB scratch

  // Pre-pass: cw = content @ cow for the whole batch (WMMA f32 16x16x4)
  cw_gemm_kernel<<<1024, 256, 0, stream>>>(text, img, cow, CW);

  // Main pass: one workgroup per sample, z DMA'd into LDS by the TDM
  const size_t smemFloats = (size_t)NMAX * ZR + 2 * ZR + 2 * ZR + 2 * NMAX +
                            DD + DD + NMAX + NMAX + 2 + 16;
  coattn_kernel<<<BB, 512, smemFloats * sizeof(float), stream>>>(
      text, img, comment, comment_num, W_ca, b_ca, W_co, b_co, CW, out);
}